// Gridnet_69750268887077
// MI455X (gfx1250) — compile-verified
//
#include <hip/hip_runtime.h>
#include <math.h>

// GridNet step, MI455X (gfx1250, wave32).
// One workgroup = one (bm,bn,bk) block x all 4 batches, resident in LDS for all
// 8 inner iterations (halos are frozen in the reference, so blocks are independent).
// Per-element 27-weight stencil => no GEMM structure => VALU kernel; CDNA5-specific
// data movement via global_load_async_to_lds_b32 (ASYNCcnt) + global_prefetch.

#define DIM   128
#define NBLK  16          // blocks per dim
#define ROWS  16          // padded k-stride of a window row (16 floats => 64B aligned rows)
#define WINF  (10*10*ROWS) // 1600 floats per window
#define WVOL  1000
#define WFLOATS (27*512)  // 13824 weight floats per block
#define SMEM_FLOATS (WFLOATS + 4*WINF + 16)

#if defined(__has_builtin)
#if __has_builtin(__builtin_amdgcn_global_load_async_to_lds_b32) && \
    __has_builtin(__builtin_amdgcn_s_wait_asynccnt)
#define USE_ASYNC_LDS 1
#endif
#endif
#ifndef USE_ASYNC_LDS
#define USE_ASYNC_LDS 0
#endif

typedef __attribute__((address_space(1))) int* gptr_t;
typedef __attribute__((address_space(3))) int* lptr_t;

__launch_bounds__(256)
__global__ void gridnet_kernel(const float* __restrict__ W,
                               const float* __restrict__ Bias,
                               const float* __restrict__ Rs,
                               const float* __restrict__ X,
                               float* __restrict__ Out)
{
    extern __shared__ float smem[];
    float* s_w    = smem;                    // 13824 floats: w[o][m][n][k] for this block
    float* s_win  = smem + WFLOATS;          // 4 windows of 10*10*ROWS
    float* s_red  = s_win + 4*WINF;          // [b*2+{sum,sumsq}] running totals
    float* s_halo = s_red + 8;               // [b*2+{sum,sumsq}] frozen halo part

    const int tid  = threadIdx.x;
    const int b    = tid >> 6;               // batch 0..3
    const int line = tid & 63;               // k-line id within block
    const int m    = line >> 3;
    const int n    = line & 7;
    const int lane = tid & 31;               // wave32 lane

    const int blk = blockIdx.x;
    const int bk  = blk & (NBLK-1);
    const int bn  = (blk >> 4) & (NBLK-1);
    const int bm  = blk >> 8;
    const int gm0 = bm << 3, gn0 = bn << 3, gk0 = bk << 3;

    // zero the 16 stat floats (s_red[0..7] + s_halo[0..7] are contiguous)
    if (tid < 16) s_red[tid] = 0.0f;

    const size_t lineoff = ((size_t)(gm0 + m)*DIM + (gn0 + n))*DIM + gk0;
    // warm L2 for the bias/rs lines we load below (gfx1250 global_prefetch_b8)
    __builtin_prefetch(Bias + lineoff, 0, 1);
    __builtin_prefetch(Rs   + lineoff, 0, 1);

    // ---- Phase 1: window fill (per batch sub-group of 64 threads) ----------
    const float* xb = X + (size_t)b * (DIM*DIM*DIM);
    for (int e = line; e < WVOL; e += 64) {
        int wm  = e / 100;
        int rem = e - wm*100;
        int wn  = rem / 10;
        int wk  = rem - wn*10;
        int gm = gm0 + wm - 1, gn = gn0 + wn - 1, gk = gk0 + wk - 1;
        bool valid = ((unsigned)gm < DIM) & ((unsigned)gn < DIM) & ((unsigned)gk < DIM);
        float* dst = &s_win[b*WINF + (wm*10 + wn)*ROWS + wk];
#if USE_ASYNC_LDS
        if (valid) {
            const float* src = &xb[((size_t)gm*DIM + gn)*DIM + gk];
            __builtin_amdgcn_global_load_async_to_lds_b32(
                (gptr_t)src, (lptr_t)dst, 0, 0);
        } else {
            *dst = 0.0f;
        }
#else
        *dst = valid ? xb[((size_t)gm*DIM + gn)*DIM + gk] : 0.0f;
#endif
    }

    // ---- Phase 2: weights -> LDS, vectorized float4 ------------------------
    const size_t wblk = ((size_t)gm0*DIM + gn0)*DIM + gk0;
    for (int v = tid; v < (WFLOATS >> 2); v += 256) {
        int o  = v >> 7;                 // 128 float4 per 512-float tile
        int r  = (v & 127) << 2;         // float index within tile
        int lm = r >> 6, ln = (r >> 3) & 7, lk = r & 7;
        float4 wv = *reinterpret_cast<const float4*>(
            W + (size_t)o*(DIM*DIM*DIM) + wblk + ((size_t)lm*DIM + ln)*DIM + lk);
        *reinterpret_cast<float4*>(&s_w[(o << 9) + (lm << 6) + (ln << 3) + lk]) = wv;
    }

    // per-thread params (bias/rs identical across batches; redundant load is fine)
    float bia[8], rsc[8];
    {
        float4 b0 = *reinterpret_cast<const float4*>(Bias + lineoff);
        float4 b1 = *reinterpret_cast<const float4*>(Bias + lineoff + 4);
        float4 r0 = *reinterpret_cast<const float4*>(Rs + lineoff);
        float4 r1 = *reinterpret_cast<const float4*>(Rs + lineoff + 4);
        bia[0]=b0.x; bia[1]=b0.y; bia[2]=b0.z; bia[3]=b0.w;
        bia[4]=b1.x; bia[5]=b1.y; bia[6]=b1.z; bia[7]=b1.w;
        rsc[0]=r0.x; rsc[1]=r0.y; rsc[2]=r0.z; rsc[3]=r0.w;
        rsc[4]=r1.x; rsc[5]=r1.y; rsc[6]=r1.z; rsc[7]=r1.w;
    }

#if USE_ASYNC_LDS
    __builtin_amdgcn_s_wait_asynccnt(0);
#endif
    __syncthreads();

    // own interior k-line, kept in registers across iterations
    float cur[8];
    {
        const float* rp = &s_win[b*WINF + ((m+1)*10 + (n+1))*ROWS + 1];
        #pragma unroll
        for (int k = 0; k < 8; ++k) cur[k] = rp[k];
    }

    // ---- Phase 3: one-time reductions --------------------------------------
    // total window sums (iteration-0 stats) and frozen halo sums
    {
        float tot = 0.f, tsq = 0.f, hal = 0.f, hsq = 0.f;
        for (int e = line; e < WVOL; e += 64) {
            int wm  = e / 100;
            int rem = e - wm*100;
            int wn  = rem / 10;
            int wk  = rem - wn*10;
            float v = s_win[b*WINF + (wm*10 + wn)*ROWS + wk];
            tot += v; tsq = fmaf(v, v, tsq);
            bool inter = ((unsigned)(wm-1) < 8u) & ((unsigned)(wn-1) < 8u) & ((unsigned)(wk-1) < 8u);
            if (!inter) { hal += v; hsq = fmaf(v, v, hsq); }
        }
        #pragma unroll
        for (int off = 16; off; off >>= 1) {
            tot += __shfl_xor(tot, off); tsq += __shfl_xor(tsq, off);
            hal += __shfl_xor(hal, off); hsq += __shfl_xor(hsq, off);
        }
        if (lane == 0) {
            atomicAdd(&s_red[2*b],    tot);  atomicAdd(&s_red[2*b+1],  tsq);
            atomicAdd(&s_halo[2*b],   hal);  atomicAdd(&s_halo[2*b+1], hsq);
        }
    }

    // wsum[e] = sum_o w[o,e]   (normalization is linear: S w(a-mu)inv = inv(S wa - mu*wsum))
    float wsum[8] = {0,0,0,0,0,0,0,0};
    for (int o = 0; o < 27; ++o) {
        float4 a = *reinterpret_cast<const float4*>(&s_w[(o << 9) + (line << 3)]);
        float4 c = *reinterpret_cast<const float4*>(&s_w[(o << 9) + (line << 3) + 4]);
        wsum[0]+=a.x; wsum[1]+=a.y; wsum[2]+=a.z; wsum[3]+=a.w;
        wsum[4]+=c.x; wsum[5]+=c.y; wsum[6]+=c.z; wsum[7]+=c.w;
    }
    __syncthreads();   // stat atomics visible before iteration 0

    // ---- Phase 4: 8 inner iterations ---------------------------------------
    const float invN = 1.0f / 1000.0f;
    for (int it = 0; it < 8; ++it) {
        float mu  = s_red[2*b]   * invN;
        float ex2 = s_red[2*b+1] * invN;
        float inv = rsqrtf((ex2 - mu*mu) + 1e-5f);
        __syncthreads();                         // everyone consumed stats
        if (line == 0) {                         // reset running sums to frozen halo part
            s_red[2*b]   = s_halo[2*b];
            s_red[2*b+1] = s_halo[2*b+1];
        }

        float acc[8] = {0,0,0,0,0,0,0,0};
        #pragma unroll
        for (int i3 = 0; i3 < 3; ++i3) {
            #pragma unroll
            for (int j3 = 0; j3 < 3; ++j3) {
                const float* rp = &s_win[b*WINF + ((m + i3)*10 + (n + j3))*ROWS];
                float4 r0 = *reinterpret_cast<const float4*>(rp);
                float4 r1 = *reinterpret_cast<const float4*>(rp + 4);
                float2 r2 = *reinterpret_cast<const float2*>(rp + 8);
                float row[10] = {r0.x,r0.y,r0.z,r0.w, r1.x,r1.y,r1.z,r1.w, r2.x,r2.y};
                const int ob = (i3*3 + j3)*3;
                #pragma unroll
                for (int k3 = 0; k3 < 3; ++k3) {
                    const float* wp = &s_w[((ob + k3) << 9) + (line << 3)];
                    float4 w0 = *reinterpret_cast<const float4*>(wp);
                    float4 w1 = *reinterpret_cast<const float4*>(wp + 4);
                    float wv[8] = {w0.x,w0.y,w0.z,w0.w, w1.x,w1.y,w1.z,w1.w};
                    #pragma unroll
                    for (int k = 0; k < 8; ++k)
                        acc[k] = fmaf(wv[k], row[k + k3], acc[k]);
                }
            }
        }
        __syncthreads();                         // all stencil reads done; reset visible

        float nsum = 0.f, nsq = 0.f;
        float* rowp = &s_win[b*WINF + ((m+1)*10 + (n+1))*ROWS + 1];
        #pragma unroll
        for (int k = 0; k < 8; ++k) {
            float a = fmaf(inv, acc[k] - mu*wsum[k], bia[k]);   // inv*(raw - mu*wsum) + bias
            float s = a / (1.0f + __expf(-a));                  // silu
            cur[k]  = fmaf(rsc[k], s, cur[k]);
            rowp[k] = cur[k];
            nsum += cur[k];
            nsq   = fmaf(cur[k], cur[k], nsq);
        }
        #pragma unroll
        for (int off = 16; off; off >>= 1) {
            nsum += __shfl_xor(nsum, off);
            nsq  += __shfl_xor(nsq,  off);
        }
        if (lane == 0) {
            atomicAdd(&s_red[2*b],   nsum);
            atomicAdd(&s_red[2*b+1], nsq);
        }
        __syncthreads();                         // writes + stat atomics done
    }

    // ---- Phase 5: write out ------------------------------------------------
    float* op = Out + (size_t)b*(DIM*DIM*DIM) + lineoff;
    *reinterpret_cast<float4*>(op)     = make_float4(cur[0], cur[1], cur[2], cur[3]);
    *reinterpret_cast<float4*>(op + 4) = make_float4(cur[4], cur[5], cur[6], cur[7]);
}

extern "C" void kernel_launch(void* const* d_in, const int* in_sizes, int n_in,
                              void* d_out, int out_size, void* d_ws, size_t ws_size,
                              hipStream_t stream) {
    (void)in_sizes; (void)n_in; (void)out_size; (void)d_ws; (void)ws_size;
    const float* W    = (const float*)d_in[0];   // (27,128,128,128)
    const float* Bias = (const float*)d_in[1];   // (128,128,128)
    const float* Rs   = (const float*)d_in[2];   // (128,128,128)
    const float* X    = (const float*)d_in[3];   // (4,128,128,128)
    // d_in[4]=inner_iterations(8), d_in[5]=block_size(8): fixed by setup, compiled in.
    float* Out = (float*)d_out;

    const size_t shmem = (size_t)SMEM_FLOATS * sizeof(float);  // ~81 KB (<320 KB WGP LDS)
    (void)hipFuncSetAttribute(reinterpret_cast<const void*>(gridnet_kernel),
                              hipFuncAttributeMaxDynamicSharedMemorySize, (int)shmem);
    gridnet_kernel<<<dim3(NBLK*NBLK*NBLK), dim3(256), shmem, stream>>>(W, Bias, Rs, X, Out);
}